// FedProtoModel_49993419325594
// MI455X (gfx1250) — compile-verified
//
#include <hip/hip_runtime.h>
#include <stdint.h>

typedef float v2f __attribute__((ext_vector_type(2)));
typedef float v8f __attribute__((ext_vector_type(8)));

#define BATCH    16384
#define K_DIM    1024
#define D_DIM    256
#define C_CLS    100
#define BM       32       // rows of x per workgroup
#define KC       256      // K-chunk staged in LDS per step
#define NCHUNK   (K_DIM / KC)
#define NTHREADS 256      // 8 wave32 waves

#define WMMA_F32(a, b, c) \
    __builtin_amdgcn_wmma_f32_16x16x4_f32(false, (a), false, (b), (short)0, (c), false, false)

// Async copy of one 32x256 f32 chunk of x into LDS (VGPR-bypassing,
// tracked by ASYNCcnt).  Each of 256 threads issues 8 b128 transfers.
__device__ __forceinline__ void stage_chunk(const float* __restrict__ x,
                                            int row0, int chunk,
                                            float* dst, int tid)
{
    #pragma unroll
    for (int i = 0; i < (BM * (KC / 4)) / NTHREADS; ++i) {   // 8 iterations
        const int v  = tid + i * NTHREADS;
        const int r  = v >> 6;             // 0..31  (KC/4 == 64 vectors per row)
        const int c4 = (v & 63) << 2;      // float column, multiple of 4 (16B)
        // LDS byte address = low 32 bits of the generic pointer (ISA aperture rule)
        const uint32_t ldsoff = (uint32_t)(uintptr_t)(dst + r * KC + c4);
        const unsigned long long ga =
            (unsigned long long)(uintptr_t)(x + (size_t)(row0 + r) * K_DIM + chunk * KC + c4);
        asm volatile("global_load_async_to_lds_b128 %0, %1, off"
                     :: "v"(ldsoff), "v"(ga) : "memory");
    }
}

__device__ __forceinline__ void wait_async0()
{
    asm volatile("s_wait_asynccnt 0x0" ::: "memory");
}

__launch_bounds__(NTHREADS)
__global__ void fedproto_fused_kernel(const float* __restrict__ x,
                                      const float* __restrict__ W,
                                      const float* __restrict__ bvec,
                                      const float* __restrict__ P,
                                      float* __restrict__ out)
{
    __shared__ __align__(16) float xs[2][BM * KC];  // 2 x 32KB double buffer
    __shared__ float zs[BM * D_DIM];                // 32KB: Z tile (32 x 256)
    __shared__ float red[BM * 8];                   // row-norm partials
    __shared__ float sumsq[BM];                     // |z|^2 per row
    __shared__ float psq[112];                      // |p|^2 per class (7*16 pad)

    const int tid  = threadIdx.x;
    const int wave = tid >> 5;
    const int lane = tid & 31;
    const int half = lane >> 4;        // ISA A/B fragment layout: K sub-offset
    const int l16  = lane & 15;
    const int row0 = blockIdx.x * BM;

    // ---- kick off async staging of chunk 0 while we compute |p|^2 ----
    stage_chunk(x, row0, 0, xs[0], tid);

    if (tid < 112) {
        float s = 0.0f;
        if (tid < C_CLS) {
            const float* pr = P + (size_t)tid * D_DIM;
            #pragma unroll 8
            for (int k = 0; k < D_DIM; ++k) s += pr[k] * pr[k];
        }
        psq[tid] = s;
    }

    wait_async0();
    __syncthreads();

    // ---- phase 1: Z = x @ W ----
    // Wave owns 32 N-columns (n0..n0+31) x all 32 rows: 4 accumulator tiles.
    v8f acc00 = {}, acc01 = {}, acc10 = {}, acc11 = {};
    const int n0   = wave * 32;
    const int arow = l16;              // A row (within 16-row tile) for this lane
    const int koff = half * 2;         // lane K sub-offset

    for (int c = 0; c < NCHUNK; ++c) {
        // prefetch next chunk into the other buffer (overlaps compute)
        if (c + 1 < NCHUNK) stage_chunk(x, row0, c + 1, xs[(c + 1) & 1], tid);

        const float* xsb  = xs[c & 1];
        const int    kb   = c * KC;

        #pragma unroll 4
        for (int kk = 0; kk < KC; kk += 4) {
            v2f a0 = *(const v2f*)&xsb[arow * KC + kk + koff];
            v2f a1 = *(const v2f*)&xsb[(arow + 16) * KC + kk + koff];
            const float* wp = W + (size_t)(kb + kk + koff) * D_DIM;
            v2f b0, b1;
            b0.x = wp[n0 + l16];        b0.y = wp[D_DIM + n0 + l16];
            b1.x = wp[n0 + 16 + l16];   b1.y = wp[D_DIM + n0 + 16 + l16];
            acc00 = WMMA_F32(a0, b0, acc00);
            acc01 = WMMA_F32(a0, b1, acc01);
            acc10 = WMMA_F32(a1, b0, acc10);
            acc11 = WMMA_F32(a1, b1, acc11);
        }

        wait_async0();      // next chunk fully in LDS
        __syncthreads();    // everyone done reading current buffer
    }

    // ---- write Z (+bias) to LDS.  C/D layout: lane l, vgpr r ->
    //      M = r + (l/16)*8, N = l%16 ----
    {
        const float bias0 = bvec[n0 + l16];
        const float bias1 = bvec[n0 + 16 + l16];
        #pragma unroll
        for (int r = 0; r < 8; ++r) {
            const int m = r + half * 8;
            zs[m * D_DIM + n0 + l16]             = acc00[r] + bias0;
            zs[m * D_DIM + n0 + 16 + l16]        = acc01[r] + bias1;
            zs[(m + 16) * D_DIM + n0 + l16]      = acc10[r] + bias0;
            zs[(m + 16) * D_DIM + n0 + 16 + l16] = acc11[r] + bias1;
        }
    }
    __syncthreads();

    // ---- row squared-norms: 8 threads per row, 32 elements each ----
    {
        const int rrow = tid >> 3;     // 0..31
        const int seg  = tid & 7;
        float s = 0.0f;
        #pragma unroll
        for (int k = 0; k < 32; ++k) {
            const float z = zs[rrow * D_DIM + seg * 32 + k];
            s += z * z;
        }
        red[rrow * 8 + seg] = s;
    }
    __syncthreads();
    if (tid < BM) {
        float s = 0.0f;
        #pragma unroll
        for (int k = 0; k < 8; ++k) s += red[tid * 8 + k];
        sumsq[tid] = s;
    }
    __syncthreads();

    // ---- phase 2: dot = Z @ P^T ; 14 tile jobs (2 row-tiles x 7 class-tiles)
    //      strided over the 8 waves ----
    for (int job = wave; job < 14; job += 8) {
        const int rowtile = job / 7;          // 0 or 1
        const int ctile   = job % 7;
        v8f accp = {};
        const int ccol = ctile * 16 + l16;
        // Clamp OOB classes to 99: those output columns are never stored,
        // and EXEC stays all-1s through the WMMAs (ISA requirement).
        const int cc = (ccol < C_CLS) ? ccol : (C_CLS - 1);
        const float* prow = P + (size_t)cc * D_DIM;      // B[k][c] = P[c][k]
        const int zrow = rowtile * 16 + arow;

        #pragma unroll 4
        for (int kk = 0; kk < D_DIM; kk += 4) {
            v2f a = *(const v2f*)&zs[zrow * D_DIM + kk + koff];
            v2f b = *(const v2f*)&prow[kk + koff];       // contiguous b64 load
            accp = WMMA_F32(a, b, accp);
        }

        if (ccol < C_CLS) {
            const float inv_d = 1.0f / (float)D_DIM;
            const float pc = psq[ccol];
            #pragma unroll
            for (int r = 0; r < 8; ++r) {
                const int m = rowtile * 16 + r + half * 8;
                const float sq = sumsq[m] - 2.0f * accp[r] + pc;
                out[(size_t)(row0 + m) * C_CLS + ccol] = -sq * inv_d;
            }
        }
    }
}

extern "C" void kernel_launch(void* const* d_in, const int* in_sizes, int n_in,
                              void* d_out, int out_size, void* d_ws, size_t ws_size,
                              hipStream_t stream) {
    (void)in_sizes; (void)n_in; (void)d_ws; (void)ws_size; (void)out_size;
    const float* x    = (const float*)d_in[0];
    const float* W    = (const float*)d_in[1];
    const float* bvec = (const float*)d_in[2];
    const float* P    = (const float*)d_in[3];
    float* out        = (float*)d_out;

    dim3 grid(BATCH / BM);       // 512 workgroups
    dim3 block(NTHREADS);        // 8 wave32 waves
    fedproto_fused_kernel<<<grid, block, 0, stream>>>(x, W, bvec, P, out);
}